// GCN_6828998000937
// MI455X (gfx1250) — compile-verified
//
#include <hip/hip_runtime.h>
#include <math.h>

#define N_NODES 50000
#define N_EDGES 1600000
#define IN_DIM  256
#define HID_DIM 128
#define OUT_DIM 40

typedef float v2f __attribute__((ext_vector_type(2)));
typedef float v8f __attribute__((ext_vector_type(8)));

// ---------------------------------------------------------------- degrees ---
__global__ __launch_bounds__(256) void k_init_deg(float* __restrict__ deg) {
    int i = blockIdx.x * 256 + threadIdx.x;
    if (i < N_NODES) deg[i] = 1.0f;   // self loop
}

__global__ __launch_bounds__(256) void k_accum_deg(const long long* __restrict__ ei,
                                                   float* __restrict__ deg) {
    int e = blockIdx.x * 256 + threadIdx.x;
    if (e < N_EDGES) {
        int d = (int)ei[N_EDGES + e];
        unsafeAtomicAdd(&deg[d], 1.0f);
    }
}

__global__ __launch_bounds__(256) void k_rsqrt_deg(float* __restrict__ deg) {
    int i = blockIdx.x * 256 + threadIdx.x;
    if (i < N_NODES) deg[i] = rsqrtf(deg[i]);
}

// ------------------------------------------------------------- WMMA GEMM1 ---
// h1[N,128] = x[N,256] @ W1[256,128]; one wave per 16x16 output tile.
// f32 16x16x4 A layout: lanes 0-15 hold M=lane, K=k..k+1; lanes 16-31 K=k+2..k+3.
__global__ __launch_bounds__(256) void k_gemm1(const float* __restrict__ X,
                                               const float* __restrict__ W,
                                               float* __restrict__ H) {
    const int lane = threadIdx.x & 31;
    const int wave = threadIdx.x >> 5;
    const int tile = blockIdx.x * 8 + wave;            // 3125*8 = 25000 tiles
    if (tile >= (N_NODES / 16) * (HID_DIM / 16)) return;
    const int nT = tile & 7;
    const int mT = tile >> 3;
    const int mr = lane & 15;
    const int kh = lane >> 4;                          // K-pair select
    const float* xrow = X + (size_t)(mT * 16 + mr) * IN_DIM;
    const float* wcol = W + nT * 16 + mr;              // column = lane&15
    v8f acc = {};
#pragma unroll 4
    for (int k = 0; k < IN_DIM; k += 4) {
        const int ka = k + 2 * kh;
        v2f a = *(const v2f*)(xrow + ka);              // 8B aligned (ka even)
        v2f b;
        b.x = wcol[(size_t)(ka + 0) * HID_DIM];
        b.y = wcol[(size_t)(ka + 1) * HID_DIM];
        acc = __builtin_amdgcn_wmma_f32_16x16x4_f32(false, a, false, b,
                                                    (short)0, acc, false, false);
    }
    // C/D layout: VGPR j -> M = j (lanes 0-15) or 8+j (lanes 16-31), N = lane&15
    float* hp = H + (size_t)(mT * 16 + 8 * kh) * HID_DIM + nT * 16 + mr;
#pragma unroll
    for (int j = 0; j < 8; ++j)
        hp[(size_t)j * HID_DIM] = acc[j];
}

// ------------------------------------------------------------- WMMA GEMM2 ---
// h2[N,40] = hr[N,128] @ W2[128,40]; N padded to 48 cols (3 tiles), guarded.
__global__ __launch_bounds__(256) void k_gemm2(const float* __restrict__ Hin,
                                               const float* __restrict__ W,
                                               float* __restrict__ Hout) {
    const int lane = threadIdx.x & 31;
    const int wave = threadIdx.x >> 5;
    const int tile = blockIdx.x * 8 + wave;            // 3125*3 = 9375 tiles
    if (tile >= (N_NODES / 16) * 3) return;
    const int nT = tile % 3;
    const int mT = tile / 3;
    const int mr = lane & 15;
    const int kh = lane >> 4;
    const int ncol = nT * 16 + mr;
    const bool nvalid = (ncol < OUT_DIM);
    const float* xrow = Hin + (size_t)(mT * 16 + mr) * HID_DIM;
    v8f acc = {};
#pragma unroll 4
    for (int k = 0; k < HID_DIM; k += 4) {
        const int ka = k + 2 * kh;
        v2f a = *(const v2f*)(xrow + ka);
        v2f b;
        b.x = nvalid ? W[(size_t)(ka + 0) * OUT_DIM + ncol] : 0.0f;
        b.y = nvalid ? W[(size_t)(ka + 1) * OUT_DIM + ncol] : 0.0f;
        acc = __builtin_amdgcn_wmma_f32_16x16x4_f32(false, a, false, b,
                                                    (short)0, acc, false, false);
    }
    if (nvalid) {
        float* hp = Hout + (size_t)(mT * 16 + 8 * kh) * OUT_DIM + ncol;
#pragma unroll
        for (int j = 0; j < 8; ++j)
            hp[(size_t)j * OUT_DIM] = acc[j];
    }
}

// -------------------------------------------------- self-loop init of agg ---
template <int DIM>
__global__ __launch_bounds__(256) void k_scale_self(const float* __restrict__ h,
                                                    const float* __restrict__ dinv,
                                                    float* __restrict__ agg) {
    int i = blockIdx.x * 256 + threadIdx.x;
    if (i < N_NODES * DIM) {
        int node = i / DIM;
        float dv = dinv[node];
        agg[i] = h[i] * dv * dv;
    }
}

// ------------------------------------------------------- edge scatter-add ---
// layer 1: wave per edge, lane moves a float4 chunk of 128 features.
__global__ __launch_bounds__(256) void k_scatter1(const long long* __restrict__ ei,
                                                  const float* __restrict__ dinv,
                                                  const float* __restrict__ h,
                                                  float* __restrict__ agg) {
    const int lane = threadIdx.x & 31;
    const int wave = threadIdx.x >> 5;
    const int e = blockIdx.x * 8 + wave;
    if (e >= N_EDGES) return;
    const int s = (int)ei[e];
    const int d = (int)ei[N_EDGES + e];
    const float coef = dinv[s] * dinv[d];
    const float4 v = *(const float4*)(h + (size_t)s * HID_DIM + lane * 4);
    float* outp = agg + (size_t)d * HID_DIM + lane * 4;
    unsafeAtomicAdd(outp + 0, v.x * coef);
    unsafeAtomicAdd(outp + 1, v.y * coef);
    unsafeAtomicAdd(outp + 2, v.z * coef);
    unsafeAtomicAdd(outp + 3, v.w * coef);
}

// layer 2: wave per edge, 40 features.
__global__ __launch_bounds__(256) void k_scatter2(const long long* __restrict__ ei,
                                                  const float* __restrict__ dinv,
                                                  const float* __restrict__ h,
                                                  float* __restrict__ agg) {
    const int lane = threadIdx.x & 31;
    const int wave = threadIdx.x >> 5;
    const int e = blockIdx.x * 8 + wave;
    if (e >= N_EDGES) return;
    const int s = (int)ei[e];
    const int d = (int)ei[N_EDGES + e];
    const float coef = dinv[s] * dinv[d];
    const float* src = h + (size_t)s * OUT_DIM;
    float* dst = agg + (size_t)d * OUT_DIM;
    for (int f = lane; f < OUT_DIM; f += 32)
        unsafeAtomicAdd(dst + f, src[f] * coef);
}

// ----------------------------------------------------------- bias + ReLU ----
__global__ __launch_bounds__(256) void k_bias_relu(float* __restrict__ agg,
                                                   const float* __restrict__ b) {
    int i = blockIdx.x * 256 + threadIdx.x;
    if (i < N_NODES * HID_DIM) {
        float v = agg[i] + b[i & (HID_DIM - 1)];
        agg[i] = fmaxf(v, 0.0f);
    }
}

// ------------------------------------------------- bias + log_softmax -------
// wave per node; lane covers features f=lane and f=lane+32 (40 total).
__global__ __launch_bounds__(256) void k_logsoftmax(float* __restrict__ out,
                                                    const float* __restrict__ b2) {
    const int lane = threadIdx.x & 31;
    const int wave = threadIdx.x >> 5;
    const int n = blockIdx.x * 8 + wave;
    if (n >= N_NODES) return;
    float* row = out + (size_t)n * OUT_DIM;
    const bool a1 = (lane + 32) < OUT_DIM;             // lanes 0-7
    float v0 = row[lane] + b2[lane];                   // lane < 40 always
    float v1 = a1 ? (row[lane + 32] + b2[lane + 32]) : -INFINITY;
    float m = fmaxf(v0, v1);
    for (int off = 16; off; off >>= 1) m = fmaxf(m, __shfl_xor(m, off));
    float s = __expf(v0 - m) + (a1 ? __expf(v1 - m) : 0.0f);
    for (int off = 16; off; off >>= 1) s += __shfl_xor(s, off);
    float lse = m + __logf(s);
    row[lane] = v0 - lse;
    if (a1) row[lane + 32] = v1 - lse;
}

// ------------------------------------------------------------------ launch --
extern "C" void kernel_launch(void* const* d_in, const int* in_sizes, int n_in,
                              void* d_out, int out_size, void* d_ws, size_t ws_size,
                              hipStream_t stream) {
    const float*     x  = (const float*)d_in[0];
    const long long* ei = (const long long*)d_in[1];
    const float*     W1 = (const float*)d_in[2];
    const float*     b1 = (const float*)d_in[3];
    const float*     W2 = (const float*)d_in[4];
    const float*     b2 = (const float*)d_in[5];
    float*           out = (float*)d_out;

    float* ws   = (float*)d_ws;
    float* dinv = ws;                                      // N
    float* h1   = dinv + N_NODES;                          // N*128
    float* agg1 = h1 + (size_t)N_NODES * HID_DIM;          // N*128
    float* h2   = agg1 + (size_t)N_NODES * HID_DIM;        // N*40
    // out doubles as agg2

    // normalization
    k_init_deg <<<(N_NODES + 255) / 256, 256, 0, stream>>>(dinv);
    k_accum_deg<<<(N_EDGES + 255) / 256, 256, 0, stream>>>(ei, dinv);
    k_rsqrt_deg<<<(N_NODES + 255) / 256, 256, 0, stream>>>(dinv);

    // layer 1
    k_gemm1<<<(N_NODES / 16) * (HID_DIM / 16) / 8, 256, 0, stream>>>(x, W1, h1);
    k_scale_self<HID_DIM><<<(N_NODES * HID_DIM + 255) / 256, 256, 0, stream>>>(h1, dinv, agg1);
    k_scatter1<<<N_EDGES / 8, 256, 0, stream>>>(ei, dinv, h1, agg1);
    k_bias_relu<<<(N_NODES * HID_DIM + 255) / 256, 256, 0, stream>>>(agg1, b1);

    // layer 2
    k_gemm2<<<((N_NODES / 16) * 3 + 7) / 8, 256, 0, stream>>>(agg1, W2, h2);
    k_scale_self<OUT_DIM><<<(N_NODES * OUT_DIM + 255) / 256, 256, 0, stream>>>(h2, dinv, out);
    k_scatter2<<<N_EDGES / 8, 256, 0, stream>>>(ei, dinv, h2, out);
    k_logsoftmax<<<(N_NODES + 7) / 8, 256, 0, stream>>>(out, b2);
}